// NormedMPNNBlock_62337155334682
// MI455X (gfx1250) — compile-verified
//
#include <hip/hip_runtime.h>
#include <hip/hip_bf16.h>

#define NN   100000
#define EE   1600000
#define DD   64
#define HHD  2
#define CCH  64
#define HC   128
#define EDD  16
#define GGR  64

typedef __attribute__((ext_vector_type(2))) float v2f;
typedef __attribute__((ext_vector_type(8))) float v8f;

// ---------------- init: zero scratch, out = bias ----------------
__global__ void k_init(float* mean, float* varr, float* cnt, float* deg,
                       float* lattr, float* denom, unsigned* nmax,
                       float* out, const float* __restrict__ bias) {
    int i = blockIdx.x * blockDim.x + threadIdx.x;
    if (i < GGR * DD) { mean[i] = 0.f; varr[i] = 0.f; }
    if (i < GGR)       cnt[i] = 0.f;
    if (i < NN)        deg[i] = 0.f;
    if (i < NN * EDD)  lattr[i] = 0.f;
    if (i < NN * HHD)  { denom[i] = 0.f; nmax[i] = 0u; }
    if (i < NN * CCH)  out[i] = bias[i & (CCH - 1)];
}

// ---------------- per-graph node counts (LDS histogram) ----------------
__global__ void k_count(const int* __restrict__ batch, float* cnt) {
    __shared__ float hist[GGR];
    if (threadIdx.x < GGR) hist[threadIdx.x] = 0.f;
    __syncthreads();
    int i = blockIdx.x * blockDim.x + threadIdx.x;
    if (i < NN) atomicAdd(&hist[batch[i]], 1.f);
    __syncthreads();
    if (threadIdx.x < GGR && hist[threadIdx.x] != 0.f)
        atomicAdd(&cnt[threadIdx.x], hist[threadIdx.x]);
}

// ---------------- segment sum of x (batch is sorted: run-flush) ----------------
__global__ void k_seg_sum(const float* __restrict__ x, const int* __restrict__ batch,
                          float* __restrict__ mean) {
    int d   = threadIdx.x & 63;
    int sub = threadIdx.x >> 6;              // 0..3
    int base = blockIdx.x * 64 + sub * 16;   // 16 nodes per sub-slice
    float acc = 0.f; int curg = -1;
    for (int i = 0; i < 16; ++i) {
        int node = base + i;
        if (node >= NN) break;
        int g = batch[node];
        if (g != curg) {
            if (curg >= 0) atomicAdd(&mean[curg * DD + d], acc);
            curg = g; acc = 0.f;
        }
        acc += x[(size_t)node * DD + d];
    }
    if (curg >= 0) atomicAdd(&mean[curg * DD + d], acc);
}

__global__ void k_mean_fin(float* mean, const float* __restrict__ cnt) {
    int i = blockIdx.x * blockDim.x + threadIdx.x;
    if (i >= GGR * DD) return;
    mean[i] /= fmaxf(cnt[i >> 6], 1.f);
}

// ---------------- segment sum of centered squares ----------------
__global__ void k_seg_var(const float* __restrict__ x, const int* __restrict__ batch,
                          const float* __restrict__ mean, const float* __restrict__ ms,
                          float* __restrict__ varr) {
    int d   = threadIdx.x & 63;
    int sub = threadIdx.x >> 6;
    int base = blockIdx.x * 64 + sub * 16;
    float msd = ms[d];
    float acc = 0.f; int curg = -1; float mg = 0.f;
    for (int i = 0; i < 16; ++i) {
        int node = base + i;
        if (node >= NN) break;
        int g = batch[node];
        if (g != curg) {
            if (curg >= 0) atomicAdd(&varr[curg * DD + d], acc);
            curg = g; acc = 0.f;
            mg = msd * mean[g * DD + d];
        }
        float xc = x[(size_t)node * DD + d] - mg;
        acc += xc * xc;
    }
    if (curg >= 0) atomicAdd(&varr[curg * DD + d], acc);
}

__global__ void k_rstd(float* varr, const float* __restrict__ cnt) {
    int i = blockIdx.x * blockDim.x + threadIdx.x;
    if (i >= GGR * DD) return;
    float c = fmaxf(cnt[i >> 6], 1.f);
    varr[i] = rsqrtf(varr[i] / c + 1e-5f);     // varr now holds rstd
}

// ---------------- h = relu(GraphNorm(x)), float4 vectorized ----------------
__global__ void k_norm(const float* __restrict__ x, const int* __restrict__ batch,
                       const float* __restrict__ mean, const float* __restrict__ rstd,
                       const float* __restrict__ ms, const float* __restrict__ gamma,
                       const float* __restrict__ beta, float* __restrict__ h) {
    int i = blockIdx.x * blockDim.x + threadIdx.x;   // over NN * (DD/4)
    if (i >= NN * (DD / 4)) return;
    int n = i >> 4, q = i & 15;
    int g = batch[n];
    float4 xv  = ((const float4*)x)[i];
    float4 mv  = ((const float4*)(mean + g * DD))[q];
    float4 rv  = ((const float4*)(rstd + g * DD))[q];
    float4 msv = ((const float4*)ms)[q];
    float4 gv  = ((const float4*)gamma)[q];
    float4 bv  = ((const float4*)beta)[q];
    float4 o;
    o.x = fmaxf((xv.x - msv.x * mv.x) * rv.x * gv.x + bv.x, 0.f);
    o.y = fmaxf((xv.y - msv.y * mv.y) * rv.y * gv.y + bv.y, 0.f);
    o.z = fmaxf((xv.z - msv.z * mv.z) * rv.z * gv.z + bv.z, 0.f);
    o.w = fmaxf((xv.w - msv.w * mv.w) * rv.w * gv.w + bv.w, 0.f);
    ((float4*)h)[i] = o;
}

// ---------------- node GEMM via V_WMMA_F32_16X16X4_F32 ----------------
// out[N,128] = h[N,64] @ W[64,128] + bvec; block = 8 waves, wave w -> 16-col strip.
__global__ void __launch_bounds__(256) k_gemm_wmma(const float* __restrict__ h,
                                                   const float* __restrict__ W,
                                                   const float* __restrict__ bvec,
                                                   float* __restrict__ out) {
    int wave = threadIdx.x >> 5;
    int lane = threadIdx.x & 31;
    int m  = lane & 15;      // row (A) / col (B,D) within tile
    int kh = lane >> 4;      // K half: lanes 16-31 hold K=2,3
    int row0 = blockIdx.x * 16;
    int col0 = wave * 16;
    const float* hrow = h + (size_t)(row0 + m) * DD;
    v8f acc = {};
#pragma unroll
    for (int kk = 0; kk < DD; kk += 4) {
        int k = kk + 2 * kh;
        v2f a; a.x = hrow[k];                    a.y = hrow[k + 1];
        v2f b; b.x = W[(size_t)k * HC + col0 + m];
               b.y = W[(size_t)(k + 1) * HC + col0 + m];
        acc = __builtin_amdgcn_wmma_f32_16x16x4_f32(false, a, false, b,
                                                    (short)0, acc, false, false);
    }
    float bb = bvec[col0 + m];
#pragma unroll
    for (int r = 0; r < 8; ++r)
        out[(size_t)(row0 + r + 8 * kh) * HC + col0 + m] = acc[r] + bb;
}

// ---------------- degree + loop_attr accumulation ----------------
__global__ void k_deg_loop(const int* __restrict__ ei, const float* __restrict__ eattr,
                           float* __restrict__ deg, float* __restrict__ lattr) {
    long i = (long)blockIdx.x * blockDim.x + threadIdx.x;
    if (i >= (long)EE * EDD) return;
    int e = (int)(i >> 4), j = (int)(i & 15);
    int dst = ei[EE + e];
    atomicAdd(&lattr[(size_t)dst * EDD + j], eattr[i]);
    if (j == 0) atomicAdd(&deg[dst], 1.f);
}

__global__ void k_loop_fin(const float* __restrict__ deg, float* __restrict__ lattr) {
    int i = blockIdx.x * blockDim.x + threadIdx.x;
    if (i >= NN * EDD) return;
    lattr[i] /= fmaxf(deg[i >> 4], 1.f);
}

// ---------------- per-edge attention logits + segment max ----------------
// one wave per edge; lane owns 4 contiguous channels (float4 path);
// lanes 0-15 = head 0, lanes 16-31 = head 1. W_e staged in LDS.
__global__ void __launch_bounds__(256) k_logit(
        const float* __restrict__ xl, const float* __restrict__ xr,
        const float* __restrict__ eattr, const float* __restrict__ lattr,
        const float* __restrict__ W_e, const float* __restrict__ att,
        const int* __restrict__ ei, float* __restrict__ logit,
        unsigned* __restrict__ nmax, int T) {
    __shared__ float sWe[EDD * HC];                 // 8 KB
    for (int t = threadIdx.x; t < EDD * HC; t += 256) sWe[t] = W_e[t];
    __syncthreads();

    int wid  = (blockIdx.x * blockDim.x + threadIdx.x) >> 5;
    int lane = threadIdx.x & 31;
    if (wid < T) {
        int src, dst; const float* ea;
        if (wid < EE) { src = ei[wid]; dst = ei[EE + wid]; ea = eattr + (size_t)wid * EDD; }
        else          { src = dst = wid - EE;              ea = lattr + (size_t)src * EDD; }

        // coalesced: lanes 0-15 fetch the 16 edge features, then broadcast
        float eaj = (lane < EDD) ? ea[lane] : 0.f;

        float4 xls = ((const float4*)(xl + (size_t)src * HC))[lane];
        float4 xrs = ((const float4*)(xr + (size_t)dst * HC))[lane];
        float4 ats = ((const float4*)att)[lane];

        float4 ee = make_float4(0.f, 0.f, 0.f, 0.f);
#pragma unroll
        for (int j = 0; j < EDD; ++j) {
            float aj = __shfl(eaj, j, 32);
            float4 w = ((const float4*)(sWe + j * HC))[lane];   // ds_load_b128
            ee.x += aj * w.x; ee.y += aj * w.y;
            ee.z += aj * w.z; ee.w += aj * w.w;
        }
        float4 v;
        v.x = xls.x + xrs.x + ee.x; v.y = xls.y + xrs.y + ee.y;
        v.z = xls.z + xrs.z + ee.z; v.w = xls.w + xrs.w + ee.w;
        v.x = v.x > 0.f ? v.x : 0.2f * v.x;
        v.y = v.y > 0.f ? v.y : 0.2f * v.y;
        v.z = v.z > 0.f ? v.z : 0.2f * v.z;
        v.w = v.w > 0.f ? v.w : 0.2f * v.w;
        float lsum = v.x * ats.x + v.y * ats.y + v.z * ats.z + v.w * ats.w;
#pragma unroll
        for (int o = 8; o > 0; o >>= 1) lsum += __shfl_xor(lsum, o, 16);
        if ((lane & 15) == 0) {
            int hh = lane >> 4;
            logit[(size_t)wid * HHD + hh] = lsum;
            unsigned key = __float_as_uint(lsum);
            key = (key & 0x80000000u) ? ~key : (key | 0x80000000u);  // order-preserving
            atomicMax(&nmax[dst * HHD + hh], key);
        }
    }
}

// ---------------- ev = exp(logit - max), denom accumulation ----------------
__global__ void k_ev(const int* __restrict__ ei, float* __restrict__ logit,
                     const unsigned* __restrict__ nmax, float* __restrict__ denom, int T) {
    int i = blockIdx.x * blockDim.x + threadIdx.x;
    if (i >= T * HHD) return;
    int e = i >> 1, hh = i & 1;
    int dst = (e < EE) ? ei[EE + e] : (e - EE);
    unsigned key = nmax[dst * HHD + hh];
    unsigned bits = (key & 0x80000000u) ? (key & 0x7FFFFFFFu) : ~key;
    float m = __uint_as_float(bits);
    float ev = expf(logit[i] - m);
    logit[i] = ev;                                // reuse buffer
    atomicAdd(&denom[dst * HHD + hh], ev);
}

// ---------------- message scatter with head-mean folded in ----------------
__global__ void k_scatter(const int* __restrict__ ei, const float* __restrict__ logit,
                          const float* __restrict__ denom, const float* __restrict__ xl,
                          float* __restrict__ out, int T) {
    long i = (long)blockIdx.x * blockDim.x + threadIdx.x;
    if (i >= (long)T * CCH) return;
    int e = (int)(i >> 6), c = (int)(i & 63);
    int src, dst;
    if (e < EE) { src = ei[e]; dst = ei[EE + e]; } else { src = dst = e - EE; }
    float a0 = logit[(size_t)e * HHD]     / (denom[dst * HHD]     + 1e-16f);
    float a1 = logit[(size_t)e * HHD + 1] / (denom[dst * HHD + 1] + 1e-16f);
    float val = 0.5f * (a0 * xl[(size_t)src * HC + c] +
                        a1 * xl[(size_t)src * HC + CCH + c]);
    atomicAdd(&out[(size_t)dst * CCH + c], val);
}

extern "C" void kernel_launch(void* const* d_in, const int* in_sizes, int n_in,
                              void* d_out, int out_size, void* d_ws, size_t ws_size,
                              hipStream_t stream) {
    const float* x     = (const float*)d_in[0];
    const float* eattr = (const float*)d_in[1];
    const float* gn_w  = (const float*)d_in[2];
    const float* gn_b  = (const float*)d_in[3];
    const float* gn_ms = (const float*)d_in[4];
    const float* W_l   = (const float*)d_in[5];
    const float* b_l   = (const float*)d_in[6];
    const float* W_r   = (const float*)d_in[7];
    const float* b_r   = (const float*)d_in[8];
    const float* W_e   = (const float*)d_in[9];
    const float* att   = (const float*)d_in[10];
    const float* bias  = (const float*)d_in[11];
    const int*   ei    = (const int*)d_in[12];
    const int*   batch = (const int*)d_in[13];
    float* out = (float*)d_out;

    float* ws = (float*)d_ws;
    size_t off = 0;
    auto alloc = [&](size_t n) { float* p = ws + off; off += n; return p; };
    float* mean  = alloc((size_t)GGR * DD);
    float* varr  = alloc((size_t)GGR * DD);      // becomes rstd
    float* cnt   = alloc(GGR);
    float* h     = alloc((size_t)NN * DD);
    float* xl    = alloc((size_t)NN * HC);
    float* xr    = alloc((size_t)NN * HC);
    float* deg   = alloc(NN);
    float* lattr = alloc((size_t)NN * EDD);
    float* logit = alloc((size_t)(EE + NN) * HHD);
    float* denom = alloc((size_t)NN * HHD);
    unsigned* nmax = (unsigned*)alloc((size_t)NN * HHD);

    const int T = EE + NN;
    dim3 blk(256);
    auto nb = [](long work) { return (unsigned)((work + 255) / 256); };

    k_init    <<<nb((long)NN * CCH), blk, 0, stream>>>(mean, varr, cnt, deg, lattr, denom, nmax, out, bias);
    k_count   <<<nb(NN), blk, 0, stream>>>(batch, cnt);
    k_seg_sum <<<(NN + 63) / 64, blk, 0, stream>>>(x, batch, mean);
    k_mean_fin<<<nb(GGR * DD), blk, 0, stream>>>(mean, cnt);
    k_seg_var <<<(NN + 63) / 64, blk, 0, stream>>>(x, batch, mean, gn_ms, varr);
    k_rstd    <<<nb(GGR * DD), blk, 0, stream>>>(varr, cnt);
    k_norm    <<<nb((long)NN * (DD / 4)), blk, 0, stream>>>(x, batch, mean, varr, gn_ms, gn_w, gn_b, h);
    k_gemm_wmma<<<NN / 16, blk, 0, stream>>>(h, W_l, b_l, xl);
    k_gemm_wmma<<<NN / 16, blk, 0, stream>>>(h, W_r, b_r, xr);
    k_deg_loop<<<nb((long)EE * EDD), blk, 0, stream>>>(ei, eattr, deg, lattr);
    k_loop_fin<<<nb((long)NN * EDD), blk, 0, stream>>>(deg, lattr);
    k_logit   <<<nb((long)T * 32), blk, 0, stream>>>(xl, xr, eattr, lattr, W_e, att, ei, logit, nmax, T);
    k_ev      <<<nb((long)T * HHD), blk, 0, stream>>>(ei, logit, nmax, denom, T);
    k_scatter <<<nb((long)T * CCH), blk, 0, stream>>>(ei, logit, denom, xl, out, T);
}